// EWADLoss_41901700940018
// MI455X (gfx1250) — compile-verified
//
#include <hip/hip_runtime.h>
#include <hip/hip_bf16.h>

#ifndef __has_builtin
#define __has_builtin(x) 0
#endif

// Async global->LDS path only on the device pass, and only if the toolchain
// exposes the gfx1250 async-copy builtin.  Fallback: direct float4 loads +
// global_prefetch_b8.
#if defined(__HIP_DEVICE_COMPILE__) && __has_builtin(__builtin_amdgcn_global_load_async_to_lds_b128)
#define EWAD_ASYNC 1
#else
#define EWAD_ASYNC 0
#endif

#if defined(__HIP_DEVICE_COMPILE__) && __has_builtin(__builtin_amdgcn_s_wait_asynccnt)
#define EWAD_WAIT_ASYNC(n) __builtin_amdgcn_s_wait_asynccnt(n)
#else
#define EWAD_WAIT_ASYNC(n) asm volatile("s_wait_asynccnt %0" ::"i"(n) : "memory")
#endif

#if EWAD_ASYNC
typedef int ewad_v4i __attribute__((ext_vector_type(4)));
typedef __attribute__((address_space(1))) ewad_v4i* ewad_gp;  // global v4i*
typedef __attribute__((address_space(3))) ewad_v4i* ewad_lp;  // lds v4i*
#endif

#define NT 256  // threads per block = 8 wave32

// One workgroup per token. Single pass over V with online log-sum-exp;
// all KL/JSD/entropy terms folded into 8 running accumulators.
__global__ __launch_bounds__(NT) void ewad_token_kernel(
    const float* __restrict__ logits,   // (S,V) student logits
    const float* __restrict__ t32,      // (S,V) teacher-32b logprobs
    const float* __restrict__ t14,      // (S,V) teacher-14b logprobs
    const int* __restrict__ labels,     // (S)
    const float* __restrict__ mask,     // (S)
    float* __restrict__ tok_loss,       // (S) per-token masked loss
    int V) {
  const int tid = threadIdx.x;
  const size_t base = (size_t)blockIdx.x * (size_t)V;

  // Running state: online softmax (m, s) + 6 plain sums.
  float m = -3.0e38f, s = 0.0f;
  float h32 = 0.0f, h14 = 0.0f;  // sum p*logp  (== -entropy)
  float d32 = 0.0f, d14 = 0.0f;  // sum p*student_logit
  float kpm = 0.0f, kqm = 0.0f;  // sum p*(logp - log_m)

  auto proc = [&](float x, float a, float b) {
    float p32 = __expf(a);
    float p14 = __expf(b);
    h32 = fmaf(p32, a, h32);
    h14 = fmaf(p14, b, h14);
    d32 = fmaf(p32, x, d32);
    d14 = fmaf(p14, x, d14);
    float lm = __logf(fmaf(0.5f, p32 + p14, 1e-8f));
    kpm = fmaf(p32, a - lm, kpm);
    kqm = fmaf(p14, b - lm, kqm);
    // online logsumexp: rarely-taken rescale branch keeps it at 1 exp/elem
    float d = x - m;
    if (d <= 0.0f) {
      s += __expf(d);
    } else {
      s = fmaf(s, __expf(-d), 1.0f);
      m = x;
    }
  };

#if EWAD_ASYNC
  // Triple-buffered async global->LDS staging: 3 tensors x 3 bufs x 4KB tiles.
  // Steady state keeps two tiles (6 async loads) in flight per wave.
  // Each lane async-loads and consumes only its own 16B -> no barriers, just
  // s_wait_asynccnt on this wave's in-flight count (done returns in order).
  __shared__ float4 bx[3][NT], ba[3][NT], bb[3][NT];
  const int TILE = NT * 4;  // floats per tile
  const int ntiles = V / TILE;
  const float* gx = logits + base + tid * 4;
  const float* ga = t32 + base + tid * 4;
  const float* gb = t14 + base + tid * 4;

#define EWAD_ISSUE(t, buf)                                                \
  do {                                                                    \
    size_t _o = (size_t)(t) * TILE;                                       \
    __builtin_amdgcn_global_load_async_to_lds_b128(                       \
        (ewad_gp)(gx + _o), (ewad_lp)&bx[buf][tid], 0, 0);                \
    __builtin_amdgcn_global_load_async_to_lds_b128(                       \
        (ewad_gp)(ga + _o), (ewad_lp)&ba[buf][tid], 0, 0);                \
    __builtin_amdgcn_global_load_async_to_lds_b128(                       \
        (ewad_gp)(gb + _o), (ewad_lp)&bb[buf][tid], 0, 0);                \
  } while (0)

  if (ntiles > 0) EWAD_ISSUE(0, 0);
  if (ntiles > 1) EWAD_ISSUE(1, 1);
  for (int t = 0; t < ntiles; ++t) {
    if (t + 2 < ntiles) {
      EWAD_ISSUE(t + 2, (t + 2) % 3);
      EWAD_WAIT_ASYNC(6);  // tiles t+1,t+2 in flight; tile t complete
    } else if (t + 1 < ntiles) {
      EWAD_WAIT_ASYNC(3);  // tile t+1 in flight; tile t complete
    } else {
      EWAD_WAIT_ASYNC(0);  // last tile complete
    }
    int b = t % 3;
    float4 x4 = bx[b][tid], a4 = ba[b][tid], c4 = bb[b][tid];
    proc(x4.x, a4.x, c4.x);
    proc(x4.y, a4.y, c4.y);
    proc(x4.z, a4.z, c4.z);
    proc(x4.w, a4.w, c4.w);
  }
  {  // tail (V % TILE floats), direct loads
    int tail = V - ntiles * TILE;
    if (tid * 4 < tail) {
      size_t o = (size_t)ntiles * TILE;
      float4 x4 = *(const float4*)(gx + o);
      float4 a4 = *(const float4*)(ga + o);
      float4 c4 = *(const float4*)(gb + o);
      proc(x4.x, a4.x, c4.x);
      proc(x4.y, a4.y, c4.y);
      proc(x4.z, a4.z, c4.z);
      proc(x4.w, a4.w, c4.w);
    }
  }
#else
  for (int i = tid * 4; i + 3 < V; i += NT * 4) {
    const float* px = logits + base + i;
    const float* pa = t32 + base + i;
    const float* pb = t14 + base + i;
    __builtin_prefetch(px + NT * 16, 0, 1);  // 4 iterations ahead
    __builtin_prefetch(pa + NT * 16, 0, 1);
    __builtin_prefetch(pb + NT * 16, 0, 1);
    float4 x4 = *(const float4*)px;
    float4 a4 = *(const float4*)pa;
    float4 c4 = *(const float4*)pb;
    proc(x4.x, a4.x, c4.x);
    proc(x4.y, a4.y, c4.y);
    proc(x4.z, a4.z, c4.z);
    proc(x4.w, a4.w, c4.w);
  }
#endif

  // Block tree-reduction of the 8-field accumulator (deterministic).
  struct Acc {
    float m, s, h32, h14, d32, d14, kpm, kqm;
  };
  __shared__ Acc red[NT];
  red[tid] = {m, s, h32, h14, d32, d14, kpm, kqm};
  __syncthreads();
  for (int off = NT / 2; off > 0; off >>= 1) {
    if (tid < off) {
      Acc o = red[tid + off];
      Acc a = red[tid];
      float M = fmaxf(a.m, o.m);
      a.s = fmaf(a.s, __expf(a.m - M), o.s * __expf(o.m - M));
      a.m = M;
      a.h32 += o.h32;
      a.h14 += o.h14;
      a.d32 += o.d32;
      a.d14 += o.d14;
      a.kpm += o.kpm;
      a.kqm += o.kqm;
      red[tid] = a;
    }
    __syncthreads();
  }

  if (tid == 0) {
    Acc A = red[0];
    float lse = A.m + __logf(A.s);
    float logv = __logf((float)V);
    // confidences from entropies (h = sum p*logp = -H)
    float e32 = fminf(fmaxf(-A.h32, 1e-8f), logv);
    float e14 = fminf(fmaxf(-A.h14, 1e-8f), logv);
    float c32 = 1.0f - e32 / logv;
    float c14 = 1.0f - e14 / logv;
    // 2-way softmax (tau=1) == sigmoid of difference
    float w32 = 1.0f / (1.0f + __expf(c14 - c32));
    float w14 = 1.0f - w32;
    // KL(t||s) = sum t*log t - sum t*x + lse   (sum t == 1)
    float kl32 = fmaxf(A.h32 - A.d32 + lse, 0.0f);
    float kl14 = fmaxf(A.h14 - A.d14 + lse, 0.0f);
    float kd = w32 * kl32 + w14 * kl14;
    // normalized JSD -> agreement -> gate
    float jsd = 0.5f * (A.kpm + A.kqm) * 1.4426950408889634f;  // /ln2
    jsd = fminf(fmaxf(jsd, 0.0f), 1.0f);
    float gate = 1.0f / (1.0f + __expf(-5.0f * ((1.0f - jsd) - 0.5f)));
    // cross-entropy with ignore_index = -100
    int lab = labels[blockIdx.x];
    float ce = 0.0f;
    if (lab != -100) ce = lse - logits[base + (size_t)lab];
    float pt = gate * kd + (1.0f - gate) * ce;
    tok_loss[blockIdx.x] = pt * mask[blockIdx.x];
  }
}

// Deterministic single-block final reduction: sum(loss)/max(sum(mask),1).
__global__ __launch_bounds__(NT) void ewad_final_kernel(
    const float* __restrict__ tok_loss, const float* __restrict__ mask,
    int ntok, float* __restrict__ out) {
  __shared__ float rs[NT], rm[NT];
  const int tid = threadIdx.x;
  float s = 0.0f, sm = 0.0f;
  for (int i = tid; i < ntok; i += NT) {
    s += tok_loss[i];
    sm += mask[i];
  }
  rs[tid] = s;
  rm[tid] = sm;
  __syncthreads();
  for (int off = NT / 2; off > 0; off >>= 1) {
    if (tid < off) {
      rs[tid] += rs[tid + off];
      rm[tid] += rm[tid + off];
    }
    __syncthreads();
  }
  if (tid == 0) out[0] = rs[0] / fmaxf(rm[0], 1.0f);
}

extern "C" void kernel_launch(void* const* d_in, const int* in_sizes, int n_in,
                              void* d_out, int out_size, void* d_ws,
                              size_t ws_size, hipStream_t stream) {
  (void)n_in;
  (void)out_size;
  (void)ws_size;
  const float* logits = (const float*)d_in[0];
  const float* t32 = (const float*)d_in[1];
  const float* t14 = (const float*)d_in[2];
  const int* labels = (const int*)d_in[3];
  const float* mask = (const float*)d_in[4];
  const int ntok = in_sizes[4];           // B*S
  const int V = in_sizes[0] / ntok;       // vocab
  float* tok_loss = (float*)d_ws;         // ntok floats of scratch

  ewad_token_kernel<<<dim3(ntok), dim3(NT), 0, stream>>>(
      logits, t32, t14, labels, mask, tok_loss, V);
  ewad_final_kernel<<<dim3(1), dim3(NT), 0, stream>>>(tok_loss, mask, ntok,
                                                      (float*)d_out);
}